// ISDALoss_18116172055066
// MI455X (gfx1250) — compile-verified
//
#include <hip/hip_runtime.h>
#include <math.h>

// Problem sizes (fixed by reference)
#define N_S 256   // batch
#define C_S 200   // classes
#define A_S 768   // feature dim
#define CP  208   // class dim padded to 13*16 for WMMA tiles
#define QW  13    // waves per block in heavy kernel (13 c-tiles of 16 = 208)

typedef float v2f __attribute__((ext_vector_type(2)));
typedef float v8f __attribute__((ext_vector_type(8)));

// ---------------- workspace layout (float offsets) ----------------
constexpr size_t WS_AVE = 0;                                 // [C,A] batch mean
constexpr size_t WS_D   = WS_AVE + (size_t)C_S * A_S;        // [C,A] Ave - ave
constexpr size_t WS_XC  = WS_D   + (size_t)C_S * A_S;        // [N,A] centered feats
constexpr size_t WS_WT  = WS_XC  + (size_t)N_S * A_S;        // [A,CP] w_fc^T (zero pad)
constexpr size_t WS_WTT = WS_WT  + (size_t)A_S * CP;         // [A,CP] w_fct^T
constexpr size_t WS_P   = WS_WTT + (size_t)A_S * CP;         // [N,CP] Xc @ W^T
constexpr size_t WS_PT  = WS_P   + (size_t)N_S * CP;         // [N,CP] Xc @ Wt^T
constexpr size_t WS_R   = WS_PT  + (size_t)N_S * CP;         // [C,CP] D @ W^T
constexpr size_t WS_RT  = WS_R   + (size_t)C_S * CP;         // [C,CP] D @ Wt^T
constexpr size_t WS_Q   = WS_RT  + (size_t)C_S * CP;         // [C,CP] dw^T CoV dw
constexpr size_t WS_QT  = WS_Q   + (size_t)C_S * CP;
constexpr size_t WS_V2  = WS_QT  + (size_t)C_S * CP;         // [C,CP] sum G^2
constexpr size_t WS_V2T = WS_V2  + (size_t)C_S * CP;
constexpr size_t WS_AMT = WS_V2T + (size_t)C_S * CP;         // [C]
constexpr size_t WS_C0  = WS_AMT + C_S;                      // [C] (1-w)
constexpr size_t WS_C1  = WS_C0  + C_S;                      // [C] w/amt_safe
constexpr size_t WS_C2  = WS_C1  + C_S;                      // [C] w(1-w)
constexpr size_t WS_CE  = WS_C2  + C_S;                      // [N]
constexpr size_t WS_KL  = WS_CE  + N_S;                      // [N]
// total ~1.19M floats (~4.8 MB)

__device__ __forceinline__ v8f wmma_f32_4(v2f a, v2f b, v8f c) {
  // V_WMMA_F32_16X16X4_F32 : D(16x16,f32) = A(16x4,f32) * B(4x16,f32) + C
  return __builtin_amdgcn_wmma_f32_16x16x4_f32(
      /*neg_a=*/false, a, /*neg_b=*/false, b,
      /*c_mod=*/(short)0, c, /*reuse_a=*/false, /*reuse_b=*/false);
}

// Async global -> LDS copy (CDNA5 ASYNC path), 16B per lane.
// ldsoff = 32-bit LDS byte offset, gptr = per-lane global address.
__device__ __forceinline__ void async_g2l_b128(unsigned ldsoff, const void* gptr) {
  asm volatile("global_load_async_to_lds_b128 %0, %1, off"
               :
               : "v"(ldsoff), "v"(gptr)
               : "memory");
}
__device__ __forceinline__ void wait_asynccnt0() {
  asm volatile("s_wait_asynccnt 0x0" ::: "memory");
}

// ---------------- per-class batch statistics ----------------
__global__ void k_stats(const float* __restrict__ feat, const float* __restrict__ AveOld,
                        const float* __restrict__ Amount, const int* __restrict__ tgt,
                        float* __restrict__ ws) {
  const int c = blockIdx.x;
  const int tid = threadIdx.x;
  __shared__ int s_t[N_S];
  __shared__ int s_cnt;
  if (tid == 0) s_cnt = 0;
  if (tid < N_S) s_t[tid] = tgt[tid];
  __syncthreads();
  if (tid < N_S && s_t[tid] == c) atomicAdd(&s_cnt, 1);
  __syncthreads();
  const int amt = s_cnt;
  const float amt_safe = (amt == 0) ? 1.0f : (float)amt;
  for (int a = tid; a < A_S; a += blockDim.x) {
    float s = 0.f;
    for (int n = 0; n < N_S; ++n)           // fixed order -> deterministic
      if (s_t[n] == c) s += feat[n * A_S + a];
    const float av = s / amt_safe;
    ws[WS_AVE + (size_t)c * A_S + a] = av;
    ws[WS_D   + (size_t)c * A_S + a] = AveOld[(size_t)c * A_S + a] - av;
  }
  if (tid == 0) {
    const float amtf = (float)amt;
    const float denom = amtf + Amount[c];
    const float w = (denom == 0.f) ? 0.f : amtf / denom;
    ws[WS_AMT + c] = amtf;
    ws[WS_C0 + c] = 1.f - w;
    ws[WS_C1 + c] = w / amt_safe;
    ws[WS_C2 + c] = w * (1.f - w);
  }
}

// ---------------- centered features ----------------
__global__ void k_center(const float* __restrict__ feat, const int* __restrict__ tgt,
                         float* __restrict__ ws) {
  const int n = blockIdx.x;
  const int l = tgt[n];
  for (int a = threadIdx.x; a < A_S; a += blockDim.x)
    ws[WS_XC + (size_t)n * A_S + a] =
        feat[(size_t)n * A_S + a] - ws[WS_AVE + (size_t)l * A_S + a];
}

// ---------------- transpose + zero-pad W, Wt into [A, CP] ----------------
__global__ void k_transpose(const float* __restrict__ W, const float* __restrict__ Wt,
                            float* __restrict__ ws) {
  const int idx = blockIdx.x * blockDim.x + threadIdx.x;
  if (idx >= A_S * CP) return;
  const int a = idx / CP;
  const int c = idx % CP;
  ws[WS_WT  + idx] = (c < C_S) ? W [(size_t)c * A_S + a] : 0.f;
  ws[WS_WTT + idx] = (c < C_S) ? Wt[(size_t)c * A_S + a] : 0.f;
}

// ---------------- generic WMMA fp32 GEMM: Out[m,n] = sum_a X[m,a]*WT[a,n] (+bias) ----------
// grid = (ceil(Mpad/16), 13), block = 32 (one wave per 16x16 tile).
// NOTE: A rows are loaded UNGUARDED (rows >= Mvalid must be readable memory);
// WMMA D rows depend only on their own A row, and we never store rows >= Mvalid.
__global__ void k_gemm(const float* __restrict__ X, const float* __restrict__ WT,
                       const float* __restrict__ bias, float* __restrict__ Out,
                       int ldOut, int Mvalid, int Cstore) {
  const int lane = threadIdx.x & 31;
  const int nlo  = lane & 15;
  const int koff = (lane >> 4) << 1;     // K sub-offset: 0 for lanes 0-15, 2 for 16-31
  const int m0 = blockIdx.x * 16;
  const int n0 = blockIdx.y * 16;
  const float* xrow = X + (size_t)(m0 + nlo) * A_S;
  v8f acc = {};
  for (int k = 0; k < A_S; k += 4) {
    const v2f a = *(const v2f*)(xrow + k + koff);     // global_load_b64
    v2f b;
    b.x = WT[(size_t)(k + koff)     * CP + n0 + nlo];
    b.y = WT[(size_t)(k + koff + 1) * CP + n0 + nlo];
    acc = wmma_f32_4(a, b, acc);
  }
  const int half8 = (lane >> 4) << 3;
  const int col = n0 + nlo;
#pragma unroll
  for (int i = 0; i < 8; ++i) {
    const int m = m0 + i + half8;
    if (m < Mvalid && col < Cstore) {
      float v = acc[i];
      if (bias) v += bias[col];
      Out[(size_t)m * ldOut + col] = v;
    }
  }
}

// ---------------- heavy per-class quadratic form: Q[l,c] = dw_c^T CoV[l] dw_c ----------
// One block per class; 13 waves, wave w owns c-tile [16w,16w+16).
// CoV row-panels (16x768 fp32, 48KB) staged via ASYNC global->LDS copies and
// read exactly once per class. F[a,c] = sum_b CoV[a,b]*dw[c,b] via WMMA, then
// Q[c] += sum_a dw[c,a]*F[a,c] with a lane-local accumulator.
__global__ __launch_bounds__(QW * 32) void k_q(const float* __restrict__ CoV,
                                               const float* __restrict__ Wf,
                                               const float* __restrict__ Wft,
                                               float* __restrict__ ws) {
  const int l = blockIdx.x;
  if (ws[WS_AMT + l] == 0.f) return;       // absent class: Q unused
  __shared__ __align__(16) float s_pan[16 * A_S];   // 48 KB CoV row-panel
  __shared__ float s_wl[A_S];
  __shared__ float s_wtl[A_S];
  const int tid  = threadIdx.x;
  const int lane = tid & 31;
  const int wave = tid >> 5;
  const int nlo  = lane & 15;
  const int koff = (lane >> 4) << 1;
  const int c0   = wave * 16;
  const float* WT  = ws + WS_WT;
  const float* WtT = ws + WS_WTT;
  for (int a = tid; a < A_S; a += QW * 32) {
    s_wl [a] = Wf [(size_t)l * A_S + a];
    s_wtl[a] = Wft[(size_t)l * A_S + a];
  }
  const float* covl = CoV + (size_t)l * A_S * A_S;
  float qacc = 0.f, qtacc = 0.f;
  for (int a0 = 0; a0 < A_S; a0 += 16) {
    __syncthreads();
    // stage CoV rows [a0, a0+16) via async copies: 16B per lane per issue
    {
      const float* src = covl + (size_t)a0 * A_S;
      for (int i = tid * 4; i < 16 * A_S; i += QW * 32 * 4)
        async_g2l_b128((unsigned)(uintptr_t)&s_pan[i], src + i);
    }
    // prefetch next panel into the cache hierarchy while async copy runs
    if (a0 + 16 < A_S) {
      const float* nxt = covl + (size_t)(a0 + 16) * A_S;
      for (int i = tid * 16; i < 16 * A_S; i += QW * 32 * 16)
        __builtin_prefetch(nxt + i, 0, 1);
    }
    wait_asynccnt0();
    __syncthreads();
    v8f acc = {}, acct = {};
    const float* prow = s_pan + (size_t)nlo * A_S;   // A row for this lane
    for (int b = 0; b < A_S; b += 4) {
      const int kk = b + koff;
      const v2f a = *(const v2f*)(prow + kk);        // ds_load_b64
      v2f bw, bt;
      bw.x = WT [(size_t)kk       * CP + c0 + nlo] - s_wl[kk];
      bw.y = WT [(size_t)(kk + 1) * CP + c0 + nlo] - s_wl[kk + 1];
      bt.x = WtT[(size_t)kk       * CP + c0 + nlo] - s_wtl[kk];
      bt.y = WtT[(size_t)(kk + 1) * CP + c0 + nlo] - s_wtl[kk + 1];
      acc  = wmma_f32_4(a, bw, acc);
      acct = wmma_f32_4(a, bt, acct);
    }
    const int half8 = (lane >> 4) << 3;
    const int cc = c0 + nlo;
#pragma unroll
    for (int i = 0; i < 8; ++i) {
      const int arow = a0 + i + half8;
      qacc  += acc[i]  * (WT [(size_t)arow * CP + cc] - s_wl [arow]);
      qtacc += acct[i] * (WtT[(size_t)arow * CP + cc] - s_wtl[arow]);
    }
  }
  // rows split across lane halves: combine, lanes 0..15 hold final Q[c0+lane]
  qacc  += __shfl_xor(qacc, 16, 32);
  qtacc += __shfl_xor(qtacc, 16, 32);
  if (lane < 16) {
    ws[WS_Q  + (size_t)l * CP + c0 + lane] = qacc;
    ws[WS_QT + (size_t)l * CP + c0 + lane] = qtacc;
  }
}

// ---------------- V2[l,c] = sum over batch members of class l of G^2 ----------------
__global__ void k_v2(const int* __restrict__ tgt, float* __restrict__ ws) {
  const int l = blockIdx.x;
  const int tid = threadIdx.x;
  __shared__ int s_t[N_S];
  if (tid < N_S) s_t[tid] = tgt[tid];
  __syncthreads();
  const float* P  = ws + WS_P;
  const float* Pt = ws + WS_PT;
  for (int c = tid; c < CP; c += blockDim.x) {
    float v = 0.f, vt = 0.f;
    for (int n = 0; n < N_S; ++n) {         // fixed order -> deterministic
      if (s_t[n] == l) {
        const float g  = P [(size_t)n * CP + c] - P [(size_t)n * CP + l];
        const float gt = Pt[(size_t)n * CP + c] - Pt[(size_t)n * CP + l];
        v += g * g;
        vt += gt * gt;
      }
    }
    ws[WS_V2  + (size_t)l * CP + c] = v;
    ws[WS_V2T + (size_t)l * CP + c] = vt;
  }
}

// ---------------- block reduction helpers (deterministic trees) ----------------
__device__ __forceinline__ float blk_max(float v, float* s, int tid) {
  s[tid] = v; __syncthreads();
  for (int o = 128; o > 0; o >>= 1) { if (tid < o) s[tid] = fmaxf(s[tid], s[tid + o]); __syncthreads(); }
  const float r = s[0]; __syncthreads();
  return r;
}
__device__ __forceinline__ float blk_sum(float v, float* s, int tid) {
  s[tid] = v; __syncthreads();
  for (int o = 128; o > 0; o >>= 1) { if (tid < o) s[tid] += s[tid + o]; __syncthreads(); }
  const float r = s[0]; __syncthreads();
  return r;
}

// ---------------- per-sample aug logits + CE / KL partials ----------------
__global__ void k_loss(const int* __restrict__ tgt, const float* __restrict__ ratio_p,
                       const float* __restrict__ y, float* __restrict__ ws) {
  const int n = blockIdx.x;
  const int tid = threadIdx.x;
  const int l = tgt[n];
  const float ratio = ratio_p[0];
  const float cc0 = ws[WS_C0 + l];
  const float cc1 = ws[WS_C1 + l];
  const float cc2 = ws[WS_C2 + l];
  const float Rll  = ws[WS_R  + (size_t)l * CP + l];
  const float Rtll = ws[WS_RT + (size_t)l * CP + l];
  __shared__ float s_aug[C_S];
  __shared__ float s_augt[C_S];
  __shared__ float s_red[256];
  for (int c = tid; c < C_S; c += 256) {
    const float h   = ws[WS_R  + (size_t)l * CP + c] - Rll;
    const float ht  = ws[WS_RT + (size_t)l * CP + c] - Rtll;
    const float s2  = ratio * (cc0 * ws[WS_Q  + (size_t)l * CP + c] +
                               cc1 * ws[WS_V2 + (size_t)l * CP + c] + cc2 * h * h);
    const float s2t = ratio * (cc0 * ws[WS_QT  + (size_t)l * CP + c] +
                               cc1 * ws[WS_V2T + (size_t)l * CP + c] + cc2 * ht * ht);
    const float yv = y[(size_t)n * C_S + c];
    s_aug[c]  = yv + 0.5f * s2;
    s_augt[c] = yv + 0.5f * s2t;
  }
  __syncthreads();
  float m = -__builtin_inff(), mt = -__builtin_inff();
  for (int c = tid; c < C_S; c += 256) { m = fmaxf(m, s_aug[c]); mt = fmaxf(mt, s_augt[c]); }
  const float gm  = blk_max(m,  s_red, tid);
  const float gmt = blk_max(mt, s_red, tid);
  float se = 0.f, set = 0.f;
  for (int c = tid; c < C_S; c += 256) { se += expf(s_aug[c] - gm); set += expf(s_augt[c] - gmt); }
  const float lse  = logf(blk_sum(se,  s_red, tid)) + gm;
  const float lset = logf(blk_sum(set, s_red, tid)) + gmt;
  float klp = 0.f;
  for (int c = tid; c < C_S; c += 256) {
    const float lpt = s_augt[c] - lset;
    const float lp  = s_aug[c]  - lse;
    klp += expf(lpt) * (lpt - lp);
  }
  const float kln = blk_sum(klp, s_red, tid);
  if (tid == 0) {
    ws[WS_CE + n] = -(s_aug[l] - lse);
    ws[WS_KL + n] = kln;
  }
}

// ---------------- final scalar reduction ----------------
__global__ void k_final(const float* __restrict__ ws, float* __restrict__ out) {
  __shared__ float s[256];
  const int tid = threadIdx.x;
  s[tid] = ws[WS_CE + tid]; __syncthreads();
  for (int o = 128; o > 0; o >>= 1) { if (tid < o) s[tid] += s[tid + o]; __syncthreads(); }
  const float cesum = s[0]; __syncthreads();
  s[tid] = ws[WS_KL + tid]; __syncthreads();
  for (int o = 128; o > 0; o >>= 1) { if (tid < o) s[tid] += s[tid + o]; __syncthreads(); }
  if (tid == 0)
    out[0] = cesum / (float)N_S + s[0] / (float)(N_S * C_S);
}

extern "C" void kernel_launch(void* const* d_in, const int* in_sizes, int n_in,
                              void* d_out, int out_size, void* d_ws, size_t ws_size,
                              hipStream_t stream) {
  (void)in_sizes; (void)n_in; (void)out_size; (void)ws_size;
  const float* feat   = (const float*)d_in[0];
  const float* w_fc   = (const float*)d_in[1];
  const float* b_fc   = (const float*)d_in[2];
  const float* w_fct  = (const float*)d_in[3];
  const float* CoV    = (const float*)d_in[4];
  const float* AveO   = (const float*)d_in[5];
  const float* Amount = (const float*)d_in[6];
  const int*   tgt    = (const int*)d_in[7];
  const float* ratio  = (const float*)d_in[8];
  float* out = (float*)d_out;
  float* ws  = (float*)d_ws;

  k_stats<<<dim3(C_S), dim3(256), 0, stream>>>(feat, AveO, Amount, tgt, ws);
  k_center<<<dim3(N_S), dim3(256), 0, stream>>>(feat, tgt, ws);
  k_transpose<<<dim3((A_S * CP + 255) / 256), dim3(256), 0, stream>>>(w_fc, w_fct, ws);

  // y = feat @ W^T + b  -> out[1:]
  k_gemm<<<dim3(N_S / 16, CP / 16), dim3(32), 0, stream>>>(
      feat, ws + WS_WT, b_fc, out + 1, C_S, N_S, C_S);
  // P = Xc @ W^T, Pt = Xc @ Wt^T
  k_gemm<<<dim3(N_S / 16, CP / 16), dim3(32), 0, stream>>>(
      ws + WS_XC, ws + WS_WT, nullptr, ws + WS_P, CP, N_S, CP);
  k_gemm<<<dim3(N_S / 16, CP / 16), dim3(32), 0, stream>>>(
      ws + WS_XC, ws + WS_WTT, nullptr, ws + WS_PT, CP, N_S, CP);
  // R = D @ W^T, Rt = D @ Wt^T  (M padded 208, 200 valid; padded A rows read
  // in-bounds workspace garbage and feed only unstored D rows)
  k_gemm<<<dim3(CP / 16, CP / 16), dim3(32), 0, stream>>>(
      ws + WS_D, ws + WS_WT, nullptr, ws + WS_R, CP, C_S, CP);
  k_gemm<<<dim3(CP / 16, CP / 16), dim3(32), 0, stream>>>(
      ws + WS_D, ws + WS_WTT, nullptr, ws + WS_RT, CP, C_S, CP);

  // heavy quadratic forms over CoV (per present class)
  k_q<<<dim3(C_S), dim3(QW * 32), 0, stream>>>(CoV, w_fc, w_fct, ws);

  k_v2<<<dim3(C_S), dim3(256), 0, stream>>>(tgt, ws);
  k_loss<<<dim3(N_S), dim3(256), 0, stream>>>(tgt, ratio, out + 1, ws);
  k_final<<<dim3(1), dim3(256), 0, stream>>>(ws, out);
}